// CsubjectFormer_22926535426295
// MI455X (gfx1250) — compile-verified
//
#include <hip/hip_runtime.h>
#include <math.h>

typedef float v2f __attribute__((ext_vector_type(2)));
typedef float v8f __attribute__((ext_vector_type(8)));

#define HD   16
#define NKV  512

// ---------------------------------------------------------------------------
// KV projection: kv[n, j] = f[n,:] . kv_w[j,:] + kv_b[j], scattered into
// per-head K/V buffers laid out [H][512][16] (row per key, contiguous d).
// ---------------------------------------------------------------------------
__global__ void kv_proj_kernel(const float* __restrict__ f,
                               const float* __restrict__ kv_w,
                               const float* __restrict__ kv_b,
                               float* __restrict__ kbuf,
                               float* __restrict__ vbuf,
                               int c) {
  int idx = blockIdx.x * blockDim.x + threadIdx.x;
  int twoc = 2 * c;
  if (idx >= NKV * twoc) return;
  int nrow = idx / twoc;
  int j    = idx % twoc;
  float acc = kv_b[j];
  const float* frow = f + (size_t)nrow * c;
  const float* wrow = kv_w + (size_t)j * c;
  for (int k = 0; k < c; ++k) acc += frow[k] * wrow[k];
  int jj = (j < c) ? j : (j - c);
  int h = jj / HD, d = jj % HD;
  float* dst = (j < c) ? kbuf : vbuf;
  dst[((size_t)h * NKV + nrow) * HD + d] = acc;
}

// ---------------------------------------------------------------------------
// Attention with two-pass softmax, fp32 WMMA 16x16x4, S computed TRANSPOSED
// (S^T = K * Q^T) so softmax reductions over keys are lane-local.
// One wave32 per (batch, head, 16-query tile).
//
// Layout assumptions (CDNA5 ISA 7.12.2, wave32):
//   A 16x4 f32 : lanes 0-15 M=lane, v0:K=0 v1:K=1 ; lanes 16-31 v0:K=2 v1:K=3
//   B  4x16 f32: v0: row K=0 (lanes 0-15) / K=2 (lanes 16-31), v1: K=1 / K=3
//   C/D 16x16  : vgpr v -> row v (lanes 0-15) / row v+8 (lanes 16-31), col=lane%16
// ---------------------------------------------------------------------------
__global__ void __launch_bounds__(32)
attn_kernel(const float* __restrict__ x,     // [b, n, c] (raw reinterpret of 5D)
            const float* __restrict__ kbuf,  // [H][512][16]
            const float* __restrict__ vbuf,  // [H][512][16]
            float* __restrict__ oout,        // [b, n, c]
            int n, int H) {
  const int c      = H * HD;
  const int qtiles = n / 16;
  const int tile = blockIdx.x;
  const int qt = tile % qtiles;
  const int h  = (tile / qtiles) % H;
  const int b  = tile / (qtiles * H);

  const int lane = threadIdx.x;
  const int half = lane >> 4;
  const int lp   = lane & 15;

  __shared__ float lbuf[16];   // softmax-denominator broadcast (per query row)

  // Q^T B-operand chunks (4x16 each), 1/sqrt(16) scale folded in.
  const float* qrow = x + (size_t)(b * n + qt * 16 + lp) * c + h * HD;
  v2f bQ[4];
#pragma unroll
  for (int kc = 0; kc < 4; ++kc) {
    const int d = 4 * kc + 2 * half;
    bQ[kc].x = qrow[d]     * 0.25f;
    bQ[kc].y = qrow[d + 1] * 0.25f;
  }

  const float* K = kbuf + (size_t)h * NKV * HD;
  const float* V = vbuf + (size_t)h * NKV * HD;

  // ---------------- pass 1: row max of S (lane-local over keys) ------------
  float mloc = -INFINITY;
  for (int t = 0; t < NKV / 16; ++t) {
    const float2* krow = (const float2*)(K + (size_t)(t * 16 + lp) * HD);
    if (t + 1 < NKV / 16)
      __builtin_prefetch((const void*)(K + (size_t)((t + 1) * 16 + lp) * HD), 0, 0);
    v8f s = {};
#pragma unroll
    for (int kc = 0; kc < 4; ++kc) {
      const float2 kk = krow[2 * kc + half];
      v2f aK; aK.x = kk.x; aK.y = kk.y;
      s = __builtin_amdgcn_wmma_f32_16x16x4_f32(false, aK, false, bQ[kc],
                                                (short)0, s, false, false);
    }
#pragma unroll
    for (int v = 0; v < 8; ++v) mloc = fmaxf(mloc, s[v]);
  }
  const float m = fmaxf(mloc, __shfl_xor(mloc, 16));   // combine key halves

  // ---------------- pass 2: P = exp(S - m), O = P @ V ----------------------
  v8f o = {};
  float lsum = 0.0f;
  for (int t = 0; t < NKV / 16; ++t) {
    const float2* krow = (const float2*)(K + (size_t)(t * 16 + lp) * HD);
    __builtin_prefetch((const void*)(V + (size_t)(t * 16 + lp) * HD), 0, 0);
    v8f s = {};
#pragma unroll
    for (int kc = 0; kc < 4; ++kc) {
      const float2 kk = krow[2 * kc + half];
      v2f aK; aK.x = kk.x; aK.y = kk.y;
      s = __builtin_amdgcn_wmma_f32_16x16x4_f32(false, aK, false, bQ[kc],
                                                (short)0, s, false, false);
    }
    // P^T values: pv[v] = P[q=lp][key_local = v + 8*half]
    float pv[8], psw[8];
#pragma unroll
    for (int v = 0; v < 8; ++v) {
      pv[v] = expf(s[v] - m);
      lsum += pv[v];
    }
#pragma unroll
    for (int v = 0; v < 8; ++v) psw[v] = __shfl_xor(pv[v], 16);

    // Re-pack P^T registers into A-operand chunks for O = P @ V:
    // aP[kc].{x,y} = P[q=lp][key = 4*kc + 2*half + {0,1}]
    v2f aP[4];
    aP[0].x = half ? psw[2] : pv[0];  aP[0].y = half ? psw[3] : pv[1];
    aP[1].x = half ? psw[6] : pv[4];  aP[1].y = half ? psw[7] : pv[5];
    aP[2].x = half ? pv[2]  : psw[0]; aP[2].y = half ? pv[3]  : psw[1];
    aP[3].x = half ? pv[6]  : psw[4]; aP[3].y = half ? pv[7]  : psw[5];

#pragma unroll
    for (int kc = 0; kc < 4; ++kc) {
      const int key = t * 16 + 4 * kc + 2 * half;
      v2f bV;
      bV.x = V[(size_t)key * HD + lp];
      bV.y = V[(size_t)(key + 1) * HD + lp];
      o = __builtin_amdgcn_wmma_f32_16x16x4_f32(false, aP[kc], false, bV,
                                                (short)0, o, false, false);
    }
  }
  lsum += __shfl_xor(lsum, 16);   // full denominator for query lp

  // Broadcast denominator to O's row mapping (row q = v + 8*half).
  lbuf[lp] = lsum;                // both halves write identical values
  __syncthreads();
  float* orow0 = oout + (size_t)(b * n + qt * 16) * c + h * HD;
#pragma unroll
  for (int v = 0; v < 8; ++v) {
    const int row = v + 8 * half;
    orow0[(size_t)row * c + lp] = o[v] / lbuf[row];
  }
}

// ---------------------------------------------------------------------------
// Per-token projection + residual + LayerNorm. blockDim.x == c (32..256),
// one block per token.
// ---------------------------------------------------------------------------
__global__ void proj_ln_kernel(const float* __restrict__ attn,
                               const float* __restrict__ x,
                               const float* __restrict__ proj_w,
                               const float* __restrict__ proj_b,
                               const float* __restrict__ ln_g,
                               const float* __restrict__ ln_b,
                               float* __restrict__ out) {
  const int c = blockDim.x;
  const int j = threadIdx.x;
  const size_t token = blockIdx.x;

  __shared__ float sh_o[256];
  __shared__ float r1[256];
  __shared__ float r2[256];

  sh_o[j] = attn[token * c + j];
  __syncthreads();

  float acc = proj_b[j];
  const float* w = proj_w + (size_t)j * c;
  for (int k = 0; k < c; ++k) acc += sh_o[k] * w[k];

  const float y = 0.5f * acc + 0.5f * x[token * c + j];
  r1[j] = y;
  r2[j] = y * y;
  for (int sft = c >> 1; sft > 0; sft >>= 1) {
    __syncthreads();
    if (j < sft) { r1[j] += r1[j + sft]; r2[j] += r2[j + sft]; }
  }
  __syncthreads();
  const float mu  = r1[0] / (float)c;
  const float var = r2[0] / (float)c - mu * mu;
  const float yn  = (y - mu) * rsqrtf(var + 1e-5f);
  out[token * c + j] = yn * ln_g[j] + ln_b[j];
}

// ---------------------------------------------------------------------------
extern "C" void kernel_launch(void* const* d_in, const int* in_sizes, int n_in,
                              void* d_out, int out_size, void* d_ws, size_t ws_size,
                              hipStream_t stream) {
  (void)in_sizes; (void)n_in; (void)out_size; (void)ws_size;
  static const int CS[4] = {32, 64, 128, 256};
  static const int NS[4] = {32768, 4096, 512, 64};
  const int B = 2;

  float* ws = (float*)d_ws;
  size_t off = 0;
  float* kptr[4]; float* vptr[4]; float* aptr[4];
  for (int i = 0; i < 4; ++i) {
    kptr[i] = ws + off; off += (size_t)NKV * CS[i];
    vptr[i] = ws + off; off += (size_t)NKV * CS[i];
  }
  for (int i = 0; i < 4; ++i) {
    aptr[i] = ws + off; off += (size_t)B * NS[i] * CS[i];
  }

  size_t out_off = 0;
  for (int i = 0; i < 4; ++i) {
    const int c = CS[i], n = NS[i], H = c / HD;
    const float* x    = (const float*)d_in[8 * i + 0];
    const float* f    = (const float*)d_in[8 * i + 1];
    const float* kv_w = (const float*)d_in[8 * i + 2];
    const float* kv_b = (const float*)d_in[8 * i + 3];
    const float* pw   = (const float*)d_in[8 * i + 4];
    const float* pb   = (const float*)d_in[8 * i + 5];
    const float* lg   = (const float*)d_in[8 * i + 6];
    const float* lb   = (const float*)d_in[8 * i + 7];
    float* out = (float*)d_out + out_off;

    const int total_kv = NKV * 2 * c;
    kv_proj_kernel<<<(total_kv + 255) / 256, 256, 0, stream>>>(
        f, kv_w, kv_b, kptr[i], vptr[i], c);

    const int blocks = B * H * (n / 16);
    attn_kernel<<<blocks, 32, 0, stream>>>(x, kptr[i], vptr[i], aptr[i], n, H);

    proj_ln_kernel<<<B * n, c, 0, stream>>>(aptr[i], x, pw, pb, lg, lb, out);

    out_off += (size_t)B * n * c;
  }

  // passthrough stages
  hipMemcpyAsync((float*)d_out + out_off, d_in[32], (size_t)5120 * sizeof(float),
                 hipMemcpyDeviceToDevice, stream);
  out_off += 5120;
  hipMemcpyAsync((float*)d_out + out_off, d_in[33], (size_t)640 * sizeof(float),
                 hipMemcpyDeviceToDevice, stream);
}